// PhysicsModel_53102975648221
// MI455X (gfx1250) — compile-verified
//
#include <hip/hip_runtime.h>

typedef float v2f __attribute__((ext_vector_type(2)));
typedef float v8f __attribute__((ext_vector_type(8)));

#define N_AGENTS 6144
#define TILE 16
#define NTILES (N_AGENTS / TILE)      // 384 tiles per dimension
#define WAVES_PER_BLOCK 8
#define BLOCK (WAVES_PER_BLOCK * 32)  // 256 threads = 8 wave32
#define DT 0.01f
#define SOFT 0.25f

struct Acc {
  float S[8], Px[8], Py[8];
};

// One 16x16 pair tile: a single V_WMMA_F32_16X16X4_F32 produces the softened
// r^2 tile via the K=4 contraction
//   [x_i, y_i, |p_i|^2, 1] . [-2x_j, -2y_j, 1, |p_j|^2 + SOFT] = r2_ij + SOFT
// (SOFT folded into B's K=3 slot so C can be the inline-0 operand -> no
// per-tile C register copies). VALU then computes the un-scaled LJ coefficient
// c = (2 i12 - i6)/r2s and folds the j-reduction into 3 FMAs/pair via
//   f_i = 24eps * (p_i * sum_j c - sum_j c*p_j).
// DIAG=true instantiation (diagonal tile only) masks the self-interaction.
template <bool DIAG>
__device__ __forceinline__ void tile_body(
    const float2* __restrict__ ldsP, int j0, int nloc, int hi,
    v2f a, float sig2, Acc& acc)
{
  const float2 pj = ldsP[j0 + nloc];          // one ds_load_b64 per tile
  const float xj = pj.x, yj = pj.y;

  // B operand (4x16, f32): col n = [-2x_j, -2y_j, 1, |p_j|^2 + SOFT]
  // lanes 0-15 -> K=0,1 ; lanes 16-31 -> K=2,3 ; N = lane%16
  const float q = __builtin_fmaf(xj, xj, __builtin_fmaf(yj, yj, SOFT));
  v2f b;
  b.x = hi ? 1.0f : (-2.0f * xj);             // K=2 : K=0
  b.y = hi ? q : (-2.0f * yj);                // K=3 : K=1

  // r2s tile = A*B + 0 (inline-constant C)
  v8f czero = {};
  v8f r2s = __builtin_amdgcn_wmma_f32_16x16x4_f32(
      false, a, false, b, (short)0, czero, false, false);

#pragma unroll
  for (int v = 0; v < 8; ++v) {
    // D layout: element (M = v + 8*hi, N = nloc)
    float t  = __builtin_amdgcn_rcpf(r2s[v]);     // 1/r2s
    float ir = sig2 * t;                          // (sigma/r)^2
    float i6 = ir * ir * ir;                      // (sigma/r)^6
    float u  = __builtin_fmaf(2.0f, i6, -1.0f);   // 2*i6 - 1
    float c  = (i6 * u) * t;                      // (2*i12 - i6)/r2s
    if (DIAG && (v + 8 * hi) == nloc) c = 0.0f;   // no self-interaction
    acc.S[v]  += c;
    acc.Px[v]  = __builtin_fmaf(c, xj, acc.Px[v]);
    acc.Py[v]  = __builtin_fmaf(c, yj, acc.Py[v]);
  }
}

__global__ __launch_bounds__(BLOCK) void lj_step_kernel(
    const float* __restrict__ pos, const float* __restrict__ vel,
    const float* __restrict__ eps_p, const float* __restrict__ sig_p,
    float* __restrict__ out)
{
  __shared__ float2 ldsP[N_AGENTS];            // 48 KB of the 320 KB WGP pool
  __shared__ float accS[TILE], accPx[TILE], accPy[TILE];

  const int tid = threadIdx.x;

  // Preload all positions into LDS as interleaved float2 (b64 transfers).
  const float2* pos2 = (const float2*)pos;
  for (int k = tid; k < N_AGENTS; k += BLOCK) ldsP[k] = pos2[k];
  if (tid < TILE) { accS[tid] = 0.f; accPx[tid] = 0.f; accPy[tid] = 0.f; }
  __syncthreads();

  const float eps  = eps_p[0];
  const float sig  = sig_p[0];
  const float sig2 = sig * sig;
  const float k24e = 24.0f * eps;

  const int itile = blockIdx.x;
  const int i0    = itile * TILE;
  // readfirstlane -> SGPR: makes the j-loop induction + diagonal-skip branch
  // fully scalar (s_cmp/s_cbranch instead of v_cmpx + EXEC save/restore).
  const int wave  = __builtin_amdgcn_readfirstlane(tid >> 5);
  const int lane  = tid & 31;
  const int nloc  = lane & 15;   // column within tile (B/C/D layout)
  const int hi    = lane >> 4;   // lane half

  // A operand (16x4, f32): row m = [x_i, y_i, |p_i|^2, 1]
  // lanes 0-15 -> K=0,1 ; lanes 16-31 -> K=2,3 ; M = lane%16
  const float2 pi = ldsP[i0 + nloc];
  v2f a;
  a.x = hi ? (pi.x * pi.x + pi.y * pi.y) : pi.x;  // K=2 : K=0
  a.y = hi ? 1.0f : pi.y;                          // K=3 : K=1

  Acc acc;
#pragma unroll
  for (int v = 0; v < 8; ++v) { acc.S[v] = 0.f; acc.Px[v] = 0.f; acc.Py[v] = 0.f; }

  // Hot loop: diagonal tile peeled out (scalar uniform branch), no masking
  // and no C-operand copies inside.
#pragma unroll 2
  for (int jt = wave; jt < NTILES; jt += WAVES_PER_BLOCK) {
    if (jt == itile) continue;                 // 1-in-48 tiles, scalar branch
    tile_body<false>(ldsP, jt * TILE, nloc, hi, a, sig2, acc);
  }
  if ((itile % WAVES_PER_BLOCK) == wave)       // owner wave runs diagonal tile
    tile_body<true>(ldsP, i0, nloc, hi, a, sig2, acc);

  // Column reduction: sum over the 16 lanes of each half (halves hold
  // disjoint M: lanes 0-15 -> M=v, lanes 16-31 -> M=v+8).
#pragma unroll
  for (int v = 0; v < 8; ++v) {
#pragma unroll
    for (int m = 1; m < 16; m <<= 1) {
      acc.S[v]  += __shfl_xor(acc.S[v],  m, 32);
      acc.Px[v] += __shfl_xor(acc.Px[v], m, 32);
      acc.Py[v] += __shfl_xor(acc.Py[v], m, 32);
    }
  }
  if (nloc == 0) {
#pragma unroll
    for (int v = 0; v < 8; ++v) {
      const int m = v + 8 * hi;
      atomicAdd(&accS[m],  acc.S[v]);          // ds_add_f32 across the 8 waves
      atomicAdd(&accPx[m], acc.Px[v]);
      atomicAdd(&accPy[m], acc.Py[v]);
    }
  }
  __syncthreads();

  // Semi-implicit Euler step for this block's 16 agents (24*eps applied here).
  if (tid < TILE) {
    const int i = i0 + tid;
    const float2 p = ldsP[i];
    const float Ssum = accS[tid];
    const float fx = k24e * (p.x * Ssum - accPx[tid]);
    const float fy = k24e * (p.y * Ssum - accPy[tid]);
    const float vx = vel[2 * i + 0] + DT * fx;   // MASS = 1
    const float vy = vel[2 * i + 1] + DT * fy;
    out[4 * i + 0] = p.x + DT * vx;
    out[4 * i + 1] = p.y + DT * vy;
    out[4 * i + 2] = vx;
    out[4 * i + 3] = vy;
  }
}

extern "C" void kernel_launch(void* const* d_in, const int* in_sizes, int n_in,
                              void* d_out, int out_size, void* d_ws, size_t ws_size,
                              hipStream_t stream) {
  const float* pos = (const float*)d_in[0];
  const float* vel = (const float*)d_in[1];
  const float* eps = (const float*)d_in[2];
  const float* sig = (const float*)d_in[3];
  float* outp = (float*)d_out;
  hipLaunchKernelGGL(lj_step_kernel, dim3(NTILES), dim3(BLOCK), 0, stream,
                     pos, vel, eps, sig, outp);
}